// Scalar3DHMM_24781961298347
// MI455X (gfx1250) — compile-verified
//
#include <hip/hip_runtime.h>
#include <hip/hip_bf16.h>

#define XY   16
#define Zdim 8
#define Tn   2048
#define Sn   2048            // XY*XY*Z
#define Bn   16
#define Ln   16
#define SENT 16
#define NB   256             // B*L sentences
#define NEGF (-3.4e38f)

#define CH     128           // K-columns staged per LDS buffer
#define CHP    132           // padded row stride (floats): 16B-aligned rows, conflict-free banks
#define NCHUNK (Tn / CH)     // 16

typedef float v2f __attribute__((ext_vector_type(2)));
typedef float v8f __attribute__((ext_vector_type(8)));

#if defined(__has_builtin)
#if __has_builtin(__builtin_amdgcn_global_load_async_to_lds_b128) && \
    __has_builtin(__builtin_amdgcn_s_wait_asynccnt)
#define USE_ASYNC_LDS 1
#endif
#endif

// exact parameter types per hipcc diagnostic: 'int __vector(4) __device__ *'
typedef int v4i __attribute__((vector_size(16)));
typedef __attribute__((address_space(1))) v4i gv4i;   // global (AS1)
typedef __attribute__((address_space(3))) v4i lv4i;   // LDS (AS3)

// move 16B global -> LDS (async path when available)
__device__ __forceinline__ void stage16(const float* g, float* l) {
#ifdef USE_ASYNC_LDS
    __builtin_amdgcn_global_load_async_to_lds_b128(
        (gv4i*)(unsigned long long)(const void*)g,
        (lv4i*)(unsigned)(unsigned long long)(const void*)l,  // addr[31:0] = LDS offset
        0, 0);
#else
    *(float4*)l = *(const float4*)g;
#endif
}

__device__ __forceinline__ void stage_wait() {
#ifdef USE_ASYNC_LDS
    __builtin_amdgcn_s_wait_asynccnt(0);
#endif
    __syncthreads();
}

// ---------------------------------------------------------------------------
// K1: fused 5x5 windowed log-mean over each (z,t) xy-plane.
__global__ __launch_bounds__(256) void k_window(const float* __restrict__ em,
                                                float* __restrict__ em2) {
    __shared__ float tile[XY][XY];
    const int zt = blockIdx.x;          // z*T + t
    const int z  = zt >> 11;
    const int t  = zt & (Tn - 1);
    const int tid = threadIdx.x;
    const int x = tid & 15, y = tid >> 4;
    const int s = z * 256 + y * 16 + x;
    tile[y][x] = em[(size_t)s * Tn + t];
    __syncthreads();

    float m = NEGF;
    #pragma unroll
    for (int dy = -2; dy <= 2; ++dy) {
        const int yy = min(max(y + dy, 0), 15);
        #pragma unroll
        for (int dx = -2; dx <= 2; ++dx) {
            const int xx = min(max(x + dx, 0), 15);
            m = fmaxf(m, tile[yy][xx]);
        }
    }
    float ssum = 0.0f;
    #pragma unroll
    for (int dy = -2; dy <= 2; ++dy) {
        const int yy = min(max(y + dy, 0), 15);
        #pragma unroll
        for (int dx = -2; dx <= 2; ++dx) {
            const int xx = min(max(x + dx, 0), 15);
            ssum += __expf(tile[yy][xx] - m);
        }
    }
    em2[(size_t)s * Tn + t] = m + __logf(ssum) - 3.2188758248682006f; // - log 25
}

// ---------------------------------------------------------------------------
// K2: zero token-count matrix (re-zeroed EVERY call: atomics accumulate).
__global__ void k_zero(float* __restrict__ p, int n) {
    int i = blockIdx.x * blockDim.x + threadIdx.x;
    if (i < n) p[i] = 0.0f;
}

// K3: token histogram. cnt packed as [T/4][NB][4]:
//   addr(t,n) = (t>>2)*NB*4 + n*4 + (t&3)  -> WMMA B-chunks are 8B vector loads
__global__ void k_hist(const int* __restrict__ stories, float* __restrict__ cnt) {
    int idx = blockIdx.x * blockDim.x + threadIdx.x;     // [0, B*L*SENT)
    if (idx >= Bn * Ln * SENT) return;
    const int b   = idx / (Ln * SENT);
    const int l   = (idx / SENT) % Ln;
    const int tok = stories[idx];
    const int n   = l * Bn + b;                          // e_all stored [L][B][S]
    if (tok >= 0 && tok < Tn)
        atomicAdd(&cnt[(size_t)(tok >> 2) * (NB * 4) + n * 4 + (tok & 3)], 1.0f);
}

// ---------------------------------------------------------------------------
// K4: e_all = em2[S x T] * cnt[T x NB] via V_WMMA_F32_16X16X4_F32.
// Block = one 16-row s-tile, 16 waves = 16 n-tiles. A-chunks (16 x CH) are
// staged global->LDS once per block (async-to-LDS, double buffered) and
// reused by all 16 waves -> 16x less A traffic than per-wave streaming.
// A 16x4 f32: lane m=lane&15, vgpr v holds K = v + 2*(lane>>4)
// B 4x16 f32: lane n=lane&15, vgpr v holds K = v + 2*(lane>>4)
// C   16x16 : lane n=lane&15, vgpr v holds M = v + 8*(lane>>4)
__global__ __launch_bounds__(512) void k_gemm(const float* __restrict__ em2,
                                              const float* __restrict__ cnt,
                                              float* __restrict__ eall) {
    __shared__ float bufA[2][16 * CHP];                  // 2 x 8.25 KB
    const int tid  = threadIdx.x;                        // 0..511
    const int wave = tid >> 5;                           // n-tile id
    const int lane = tid & 31;
    const int half = lane >> 4, lm = lane & 15;
    const int s0 = blockIdx.x * 16;
    const int n0 = wave * 16;

    // staging map: thread moves 16B; 32 threads cover one CH-float row
    const int arow = tid >> 5;
    const int acol = (tid & 31) * 4;
    const float* gsrc = em2 + (size_t)(s0 + arow) * Tn + acol;
    float*       ldst = &bufA[0][arow * CHP + acol];     // buffer 1 is +16*CHP

    const float* Bp = cnt + (size_t)(n0 + lm) * 4 + 2 * half;

    stage16(gsrc, ldst);                                 // chunk 0 -> buf 0
    stage_wait();

    v8f acc = {0.f, 0.f, 0.f, 0.f, 0.f, 0.f, 0.f, 0.f};
    for (int c = 0; c < NCHUNK; ++c) {
        const int buf = c & 1;
        if (c + 1 < NCHUNK)                              // prefetch next chunk
            stage16(gsrc + (size_t)(c + 1) * CH, ldst + (buf ^ 1) * (16 * CHP));
        __builtin_prefetch(Bp + (size_t)(c + 1) * (CH / 4) * (NB * 4), 0, 3);

        const float* Ar = &bufA[buf][lm * CHP + 2 * half];
        const float* Bc = Bp + (size_t)c * (CH / 4) * (NB * 4);
        #pragma unroll 4
        for (int q = 0; q < CH / 4; ++q) {
            v2f a = *(const v2f*)(Ar + q * 4);                    // ds_load_b64
            v2f b = *(const v2f*)(Bc + (size_t)q * (NB * 4));     // global b64
            acc = __builtin_amdgcn_wmma_f32_16x16x4_f32(
                      false, a, false, b, (short)0, acc, false, false);
        }
        stage_wait();                                    // async done + all waves done reading
    }

    float* o = eall + (size_t)(n0 + lm) * Sn + s0 + 8 * half;
    #pragma unroll
    for (int v = 0; v < 8; ++v) o[v] = acc[v];
}

// ---------------------------------------------------------------------------
// K5: pack the 7 transition diagonals for coalesced recursion loads.
__global__ void k_diag(const float* __restrict__ trans, float* __restrict__ diag) {
    int idx = blockIdx.x * blockDim.x + threadIdx.x;     // [0, 7*S)
    if (idx >= 7 * Sn) return;
    const int j = idx >> 11;
    const int h = idx & (Sn - 1);
    const int off[7] = {-16, -1, 0, 1, 16, 256, 512};
    const int i = h + off[j];
    diag[idx] = ((unsigned)i < Sn) ? trans[(size_t)h * Sn + i] : -1e9f;
}

// K6: out[0,b,s] = e_all[0,b,s] + priors[s]
__global__ void k_init(const float* __restrict__ eall, const float* __restrict__ priors,
                       float* __restrict__ out) {
    int idx = blockIdx.x * blockDim.x + threadIdx.x;     // [0, B*S)
    const int b = idx >> 11, s = idx & (Sn - 1);
    out[idx] = eall[(size_t)b * Sn + s] + priors[s];
}

// K7: one forward step. Sparse 7-point logsumexp (exact per-row max),
// identical to the dense reference since exp(-1e9 - m) == 0 in fp32.
__global__ void k_step(const float* __restrict__ diag, const float* __restrict__ eall,
                       float* __restrict__ out, int t) {
    int idx = blockIdx.x * blockDim.x + threadIdx.x;     // [0, B*S)
    const int b = idx >> 11, h = idx & (Sn - 1);
    const float* prev = out + (size_t)(t - 1) * Bn * Sn + (size_t)b * Sn;
    const int off[7] = {-16, -1, 0, 1, 16, 256, 512};
    float v[7];
    float m = NEGF;
    #pragma unroll
    for (int j = 0; j < 7; ++j) {
        const int i = h + off[j];
        v[j] = ((unsigned)i < Sn) ? (diag[(size_t)j * Sn + h] + prev[i]) : NEGF;
        m = fmaxf(m, v[j]);
    }
    float ssum = 0.0f;
    #pragma unroll
    for (int j = 0; j < 7; ++j)
        if (v[j] > -3.0e38f) ssum += __expf(v[j] - m);
    out[(size_t)t * Bn * Sn + idx] =
        eall[(size_t)(t * Bn + b) * Sn + h] + m + __logf(ssum);
}

// ---------------------------------------------------------------------------
extern "C" void kernel_launch(void* const* d_in, const int* in_sizes, int n_in,
                              void* d_out, int out_size, void* d_ws, size_t ws_size,
                              hipStream_t stream) {
    const int*   stories = (const int*)d_in[0];
    // d_in[1] = story_length (compile-time Ln)
    const float* priors  = (const float*)d_in[2];
    const float* trans   = (const float*)d_in[3];
    const float* emis    = (const float*)d_in[4];
    float*       out     = (float*)d_out;

    char* ws = (char*)d_ws;
    float* em2  = (float*)ws;                                       // S*T   = 16 MB
    float* cnt  = (float*)(ws + (size_t)Sn * Tn * 4);               // T*NB  =  2 MB
    float* eall = (float*)(ws + (size_t)Sn * Tn * 4
                              + (size_t)Tn * NB * 4);               // NB*S  =  2 MB
    float* diag = (float*)(ws + (size_t)Sn * Tn * 4
                              + (size_t)Tn * NB * 4
                              + (size_t)NB * Sn * 4);               // 7*S   = 56 KB

    // 1) fused 5x5 windowed log-mean
    k_window<<<Zdim * Tn, 256, 0, stream>>>(emis, em2);
    // 2) token histogram (zero + scatter)
    k_zero<<<(Tn * NB + 255) / 256, 256, 0, stream>>>(cnt, Tn * NB);
    k_hist<<<(Bn * Ln * SENT + 255) / 256, 256, 0, stream>>>(stories, cnt);
    // 3) e_all = em2 x cnt  (fp32 WMMA, LDS-staged A with async copy)
    k_gemm<<<Sn / 16, 512, 0, stream>>>(em2, cnt, eall);
    // 4) transition diagonals + init + 15 forward steps
    k_diag<<<(7 * Sn + 255) / 256, 256, 0, stream>>>(trans, diag);
    k_init<<<(Bn * Sn) / 256, 256, 0, stream>>>(eall, priors, out);
    for (int t = 1; t < Ln; ++t)
        k_step<<<(Bn * Sn) / 256, 256, 0, stream>>>(diag, eall, out, t);
}